// YOLOv1Loss_4123168604169
// MI455X (gfx1250) — compile-verified
//
#include <hip/hip_runtime.h>

typedef __attribute__((ext_vector_type(4))) float v4f;
typedef __attribute__((ext_vector_type(2))) float v2f;
typedef __attribute__((ext_vector_type(8))) float v8f;

#define CELLS 256            // cells per block tile
#define PRED_F (CELLS * 30)  // 7680 floats  = 30720 B (16B-aligned tiles)
#define TARG_F (CELLS * 25)  // 6400 floats  = 25600 B (16B-aligned tiles)

// ---------------------------------------------------------------------------
// Kernel 1: stream pred/target through LDS, compute per-cell YOLO loss terms,
// reduce to one partial per block (deterministic, no atomics).
// Bandwidth-bound: ~177 MB single pass -> ~7.6 us floor at 23.3 TB/s.
// ---------------------------------------------------------------------------
__global__ __launch_bounds__(256) void yolo_partial(const float* __restrict__ pred,
                                                    const float* __restrict__ targ,
                                                    float* __restrict__ partials,
                                                    int nCells) {
    __shared__ alignas(16) float sPred[PRED_F];
    __shared__ alignas(16) float sTarg[TARG_F];
    __shared__ float sW[8];

    const int tid   = threadIdx.x;
    const int tile  = blockIdx.x;
    const int cell0 = tile * CELLS;
    int cells = nCells - cell0;
    if (cells > CELLS) cells = CELLS;

    if (cells == CELLS) {
        // Full tile: 16B-aligned (256*30*4 and 256*25*4 are multiples of 16).
        const v4f* p4 = (const v4f*)(pred + (size_t)cell0 * 30);
        const v4f* t4 = (const v4f*)(targ + (size_t)cell0 * 25);
        v4f* sp4 = (v4f*)sPred;
        v4f* st4 = (v4f*)sTarg;
        for (int i = tid; i < PRED_F / 4; i += 256)
            sp4[i] = __builtin_nontemporal_load(&p4[i]);   // single-use stream
        for (int i = tid; i < TARG_F / 4; i += 256)
            st4[i] = __builtin_nontemporal_load(&t4[i]);
    } else {
        // Ragged tail (not hit for N=16384, kept for generality).
        for (int i = tid; i < cells * 30; i += 256) sPred[i] = pred[(size_t)cell0 * 30 + i];
        for (int i = tid; i < cells * 25; i += 256) sTarg[i] = targ[(size_t)cell0 * 25 + i];
    }
    __syncthreads();

    float acc = 0.0f;
    if (tid < cells) {
        const float* lp = sPred + tid * 30;  // word stride 30: conflict-free on 64 banks
        const float* lt = sTarg + tid * 25;  // word stride 25: conflict-free on 64 banks

        const float tc  = lt[0];
        const float p0c = lp[0];
        const float p1c = lp[5];

        if (tc == 0.0f) {
            // noobj: 0.5 * sum over both boxes of (conf - 0)^2
            float d0 = p0c - tc, d1 = p1c - tc;
            acc += 0.5f * (d0 * d0 + d1 * d1);
        }
        if (tc == 1.0f) {
            const float tx = lt[1], ty = lt[2], tw = lt[3], th = lt[4];
            const float tX0 = tx - tw * 0.5f, tY0 = ty - th * 0.5f;
            const float tX1 = tx + tw * 0.5f, tY1 = ty + th * 0.5f;
            const float areaT = (tX1 - tX0) * (tY1 - tY0);

            float iou0 = 0.0f, iou1 = 0.0f;
#pragma unroll
            for (int bb = 0; bb < 2; ++bb) {
                const float* q = lp + 5 * bb;
                const float px = q[1], py = q[2], pw = q[3], ph = q[4];
                const float pX0 = px - pw * 0.5f, pY0 = py - ph * 0.5f;
                const float pX1 = px + pw * 0.5f, pY1 = py + ph * 0.5f;
                float iw = fminf(pX1, tX1) - fmaxf(pX0, tX0);
                float ih = fminf(pY1, tY1) - fmaxf(pY0, tY0);
                iw = fmaxf(iw, 0.0f);
                ih = fmaxf(ih, 0.0f);
                const float inter = iw * ih;
                const float areaP = (pX1 - pX0) * (pY1 - pY0);
                const float iou = inter / (areaP + areaT - inter);
                if (bb == 0) iou0 = iou; else iou1 = iou;
            }
            // argmax (first index wins ties -> box0 unless iou1 strictly greater)
            const int sel  = (iou1 > iou0) ? 1 : 0;
            const float* r = lp + 5 * sel;

            const float dx = r[1] - tx, dy = r[2] - ty;
            acc += 5.0f * (dx * dx + dy * dy);                       // xy_loss
            const float dw = sqrtf(r[3]) - sqrtf(tw);
            const float dh = sqrtf(r[4]) - sqrtf(th);
            acc += 5.0f * (dw * dw + dh * dh);                       // wh_loss
            const float dc = r[0] - tc;
            acc += dc * dc;                                          // obj_conf
#pragma unroll
            for (int k = 0; k < 20; ++k) {                           // cls_loss
                const float d = lp[10 + k] - lt[5 + k];
                acc += d * d;
            }
        }
    }

    // wave32 reduction, then 8 wave partials -> block partial
    for (int off = 16; off > 0; off >>= 1)
        acc += __shfl_down(acc, off, 32);
    if ((tid & 31) == 0) sW[tid >> 5] = acc;
    __syncthreads();
    if (tid == 0) {
        float s = 0.0f;
#pragma unroll
        for (int i = 0; i < 8; ++i) s += sW[i];
        partials[tile] = s;
    }
}

// ---------------------------------------------------------------------------
// Kernel 2: one wave32 reduces all block partials with V_WMMA_F32_16X16X4_F32.
// Each 64-float chunk fills the 16x4 f32 A-matrix (2 VGPRs, contiguous lane
// pairs -> one global_load_b64 per WMMA); B = all-ones, C accumulates.
// Any bijection of the chunk onto A slots is valid since B is constant 1.
// Two independent accumulator chains hide the D->C WMMA latency.
// D[m][n] replicated over n; total = (sum of 8 D VGPRs at lane 0, rows 0-7)
//                                  + (same at lane 16, rows 8-15).
// ---------------------------------------------------------------------------
__global__ __launch_bounds__(32) void yolo_reduce(const float* __restrict__ partials,
                                                  float* __restrict__ out,
                                                  int n, float inv_n) {
    const int lane = threadIdx.x;  // 0..31, all active: EXEC all-1s for WMMA
    v2f b;
    b.x = 1.0f;
    b.y = 1.0f;
    v8f c0 = {};
    v8f c1 = {};

    const v2f* p2 = (const v2f*)partials;   // d_ws is >=8B aligned
    const int nFull = n & ~63;              // full 64-element chunks (no guards)
    int base = 0;
    for (; base + 128 <= nFull; base += 128) {
        const v2f a0 = p2[(base >> 1) + lane];          // global_load_b64
        const v2f a1 = p2[((base + 64) >> 1) + lane];   // global_load_b64
        c0 = __builtin_amdgcn_wmma_f32_16x16x4_f32(false, a0, false, b,
                                                   (short)0, c0, false, false);
        c1 = __builtin_amdgcn_wmma_f32_16x16x4_f32(false, a1, false, b,
                                                   (short)0, c1, false, false);
    }
    for (; base < nFull; base += 64) {
        const v2f a = p2[(base >> 1) + lane];
        c0 = __builtin_amdgcn_wmma_f32_16x16x4_f32(false, a, false, b,
                                                   (short)0, c0, false, false);
    }
    if (base < n) {  // ragged tail (never hit for 3136 partials), guarded
        const int i0 = base + 2 * lane;
        const int i1 = i0 + 1;
        v2f a;
        a.x = (i0 < n) ? partials[i0] : 0.0f;
        a.y = (i1 < n) ? partials[i1] : 0.0f;
        c1 = __builtin_amdgcn_wmma_f32_16x16x4_f32(false, a, false, b,
                                                   (short)0, c1, false, false);
    }

    float s = 0.0f;
#pragma unroll
    for (int i = 0; i < 8; ++i) s += c0[i] + c1[i];
    const float other = __shfl(s, 16, 32);  // rows 8..15 live at lane 16
    if (lane == 0) out[0] = (s + other) * inv_n;
}

// ---------------------------------------------------------------------------
extern "C" void kernel_launch(void* const* d_in, const int* in_sizes, int n_in,
                              void* d_out, int out_size, void* d_ws, size_t ws_size,
                              hipStream_t stream) {
    const float* pred = (const float*)d_in[0];
    const float* targ = (const float*)d_in[1];
    float* out      = (float*)d_out;
    float* partials = (float*)d_ws;

    const int predElems = in_sizes[0];          // N * 49 * 30
    const int nCells    = predElems / 30;       // N * 49 = 802816
    const int N         = nCells / 49;          // 16384
    const int blocks    = (nCells + CELLS - 1) / CELLS;  // 3136

    yolo_partial<<<blocks, 256, 0, stream>>>(pred, targ, partials, nCells);
    yolo_reduce<<<1, 32, 0, stream>>>(partials, out, blocks, 1.0f / (float)N);
}